// Decoder_14843406975297
// MI455X (gfx1250) — compile-verified
//
#include <hip/hip_runtime.h>
#include <hip/hip_bf16.h>

#define VOCAB 32000
#define EMB   512
#define HID   1024
#define SEQ   512
#define BATCH 128
#define H2    2048          // 2*HID
#define RNNIN 2560          // 2*HID + EMB
#define G4    4096          // 4*HID

typedef float v2f __attribute__((ext_vector_type(2)));
typedef float v8f __attribute__((ext_vector_type(8)));

// ---------------------------------------------------------------------------
// 1) Embedding gather: rnn_in[b, 2048:2560] = emb_table[x[b], :]
// ---------------------------------------------------------------------------
__global__ void embed_kernel(const int* __restrict__ x,
                             const float* __restrict__ emb_table,
                             float* __restrict__ rnn_in) {
    int idx = blockIdx.x * 256 + threadIdx.x;          // 0 .. 128*512-1
    int b = idx >> 9;
    int e = idx & 511;
    rnn_in[(size_t)b * RNNIN + H2 + e] = emb_table[(size_t)x[b] * EMB + e];
}

// ---------------------------------------------------------------------------
// 2) Fused attention: one pass over encoder_states with online softmax.
//    One block per batch column b; 256 threads; ctx kept in registers
//    (8 floats/thread covering the 2048-wide context row).
//    rnn_in[b, 0:2048] = context[b, :]
// ---------------------------------------------------------------------------
__global__ __launch_bounds__(256) void attn_ctx_kernel(
        const float* __restrict__ enc,     // [SEQ, BATCH, 2H]
        const float* __restrict__ hidden,  // [BATCH, H]
        const float* __restrict__ W_e,     // [3H]
        const float* __restrict__ b_e,     // [1]
        float* __restrict__ rnn_in) {      // [BATCH, RNNIN]
    __shared__ float w2[H2];
    __shared__ float red[2][8];
    __shared__ float hdot_s;

    const int b    = blockIdx.x;
    const int t    = threadIdx.x;
    const int lane = t & 31;
    const int wv   = t >> 5;

    // stage W_e[H:3H] (the encoder part) into LDS
    for (int j = t; j < H2; j += 256) w2[j] = W_e[HID + j];

    // hdot = hidden[b] . W_e[0:H] + b_e  (constant across s)
    float p = 0.f;
    const float* hrow = hidden + (size_t)b * HID;
    for (int j = t; j < HID; j += 256) p += hrow[j] * W_e[j];
    for (int off = 16; off; off >>= 1) p += __shfl_xor(p, off, 32);
    if (lane == 0) red[0][wv] = p;
    __syncthreads();                    // also orders the w2[] stores
    if (t == 0) {
        float s = 0.f;
        for (int i = 0; i < 8; ++i) s += red[0][i];
        hdot_s = s + b_e[0];
    }
    __syncthreads();
    const float hdot = hdot_s;

    float ctx[8];
    float ev[8];
#pragma unroll
    for (int u = 0; u < 8; ++u) ctx[u] = 0.f;
    float m = 0.f, l = 0.f;             // relu output >= 0, so m=0 is a valid floor

    for (int s = 0; s < SEQ; ++s) {
        const float* er = enc + ((size_t)s * BATCH + b) * H2;
        float part = 0.f;
#pragma unroll
        for (int u = 0; u < 8; ++u) {
            int j = t + 256 * u;
            float e = __builtin_nontemporal_load(er + j);   // streamed, no reuse
            ev[u] = e;
            part += e * w2[j];
        }
        for (int off = 16; off; off >>= 1) part += __shfl_xor(part, off, 32);
        if (lane == 0) red[s & 1][wv] = part;               // parity double-buffer
        __syncthreads();
        float dot = 0.f;
#pragma unroll
        for (int i = 0; i < 8; ++i) dot += red[s & 1][i];

        // online softmax update (identical scalar trajectory on every thread)
        float e   = fmaxf(hdot + dot, 0.f);                 // relu energy
        float mn  = fmaxf(m, e);
        float sc  = __expf(m - mn);
        float pe  = __expf(e - mn);
        l = l * sc + pe;
#pragma unroll
        for (int u = 0; u < 8; ++u) ctx[u] = ctx[u] * sc + pe * ev[u];
        m = mn;
    }

    const float inv = 1.f / l;
#pragma unroll
    for (int u = 0; u < 8; ++u)
        rnn_in[(size_t)b * RNNIN + t + 256 * u] = ctx[u] * inv;
}

// ---------------------------------------------------------------------------
// 3) LSTM gate GEMM via V_WMMA_F32_16X16X4_F32.
//    gates[128,4096] = rnn_in[128,2560] @ W_ih^T + hidden[128,1024] @ W_hh^T + biases
//    Grid: 4096/16 N-tiles; 8 waves/block = all 8 M-tiles of the batch.
//    A layout (32-bit 16x4): lanes 0-15 -> K=k,k+1 ; lanes 16-31 -> K=k+2,k+3.
//    B layout (4x16, per-lane column): same K split, lane&15 selects column.
// ---------------------------------------------------------------------------
__global__ __launch_bounds__(256) void gates_kernel(
        const float* __restrict__ rnn_in,  // [BATCH, RNNIN]
        const float* __restrict__ hidden,  // [BATCH, HID]
        const float* __restrict__ W_ih,    // [G4, RNNIN]
        const float* __restrict__ W_hh,    // [G4, HID]
        const float* __restrict__ b_ih,    // [G4]
        const float* __restrict__ b_hh,    // [G4]
        float* __restrict__ gates) {       // [BATCH, G4]
    const int t    = threadIdx.x;
    const int lane = t & 31;
    const int wv   = t >> 5;
    const int half = lane >> 4;
    const int r    = lane & 15;
    const int n0   = blockIdx.x * 16;
    const int m0   = wv * 16;

    v8f c = {};

    const float* arow = rnn_in + (size_t)(m0 + r) * RNNIN;
    const float* brow = W_ih   + (size_t)(n0 + r) * RNNIN;
    for (int k = 0; k < RNNIN; k += 4) {
        const int kk = k + 2 * half;
        v2f a  = *(const v2f*)(arow + kk);
        v2f bb = *(const v2f*)(brow + kk);
        c = __builtin_amdgcn_wmma_f32_16x16x4_f32(false, a, false, bb,
                                                  (short)0, c, false, false);
    }

    const float* arow2 = hidden + (size_t)(m0 + r) * HID;
    const float* brow2 = W_hh   + (size_t)(n0 + r) * HID;
    for (int k = 0; k < HID; k += 4) {
        const int kk = k + 2 * half;
        v2f a  = *(const v2f*)(arow2 + kk);
        v2f bb = *(const v2f*)(brow2 + kk);
        c = __builtin_amdgcn_wmma_f32_16x16x4_f32(false, a, false, bb,
                                                  (short)0, c, false, false);
    }

    const int n = n0 + r;
    const float bias = b_ih[n] + b_hh[n];
#pragma unroll
    for (int j = 0; j < 8; ++j) {
        const int m = m0 + j + 8 * half;          // C/D: VGPR j -> M = j + 8*half
        gates[(size_t)m * G4 + n] = c[j] + bias;
    }
}

// ---------------------------------------------------------------------------
// 4) LSTM pointwise: i,f,g,o -> c_new, h_new
// ---------------------------------------------------------------------------
__global__ void lstm_kernel(const float* __restrict__ gates,
                            const float* __restrict__ cell,   // [BATCH, HID]
                            float* __restrict__ h_new,
                            float* __restrict__ c_new) {
    int idx = blockIdx.x * 256 + threadIdx.x;     // 0 .. 128*1024-1
    int b = idx >> 10;
    int j = idx & 1023;
    const float* g = gates + (size_t)b * G4;
    float ig = 1.f / (1.f + __expf(-g[j]));
    float fg = 1.f / (1.f + __expf(-g[j + HID]));
    float gg = tanhf(g[j + 2 * HID]);
    float og = 1.f / (1.f + __expf(-g[j + 3 * HID]));
    float cn = fg * cell[idx] + ig * gg;
    c_new[idx] = cn;
    h_new[idx] = og * tanhf(cn);
}

// ---------------------------------------------------------------------------
// 5) FC head GEMM via WMMA: preds[128,32000] = h_new @ W_fc^T + b_fc
// ---------------------------------------------------------------------------
__global__ __launch_bounds__(256) void fc_kernel(
        const float* __restrict__ h,      // [BATCH, HID]
        const float* __restrict__ W_fc,   // [VOCAB, HID]
        const float* __restrict__ b_fc,   // [VOCAB]
        float* __restrict__ out) {        // [BATCH, VOCAB]
    const int t    = threadIdx.x;
    const int lane = t & 31;
    const int wv   = t >> 5;
    const int half = lane >> 4;
    const int r    = lane & 15;
    const int n0   = blockIdx.x * 16;
    const int m0   = wv * 16;

    v8f c = {};
    const float* arow = h    + (size_t)(m0 + r) * HID;
    const float* brow = W_fc + (size_t)(n0 + r) * HID;
    for (int k = 0; k < HID; k += 4) {
        const int kk = k + 2 * half;
        v2f a  = *(const v2f*)(arow + kk);
        v2f bb = *(const v2f*)(brow + kk);
        c = __builtin_amdgcn_wmma_f32_16x16x4_f32(false, a, false, bb,
                                                  (short)0, c, false, false);
    }

    const int n = n0 + r;
    const float bias = b_fc[n];
#pragma unroll
    for (int j = 0; j < 8; ++j) {
        const int m = m0 + j + 8 * half;
        out[(size_t)m * VOCAB + n] = c[j] + bias;
    }
}

// ---------------------------------------------------------------------------
// Launch
// ---------------------------------------------------------------------------
extern "C" void kernel_launch(void* const* d_in, const int* in_sizes, int n_in,
                              void* d_out, int out_size, void* d_ws, size_t ws_size,
                              hipStream_t stream) {
    const int*   x      = (const int*)  d_in[0];
    const float* enc    = (const float*)d_in[1];
    const float* hidden = (const float*)d_in[2];   // [1,128,1024]
    const float* cell   = (const float*)d_in[3];   // [1,128,1024]
    const float* embt   = (const float*)d_in[4];
    const float* W_e    = (const float*)d_in[5];
    const float* b_e    = (const float*)d_in[6];
    const float* W_ih   = (const float*)d_in[7];
    const float* W_hh   = (const float*)d_in[8];
    const float* b_ih   = (const float*)d_in[9];
    const float* b_hh   = (const float*)d_in[10];
    const float* W_fc   = (const float*)d_in[11];
    const float* b_fc   = (const float*)d_in[12];

    float* out   = (float*)d_out;
    float* preds = out;                                   // [128*32000]
    float* h_new = out + (size_t)BATCH * VOCAB;           // [128*1024]
    float* c_new = h_new + (size_t)BATCH * HID;           // [128*1024]

    float* ws     = (float*)d_ws;
    float* rnn_in = ws;                                   // [128*2560]
    float* gates  = ws + (size_t)BATCH * RNNIN;           // [128*4096]

    embed_kernel   <<<(BATCH * EMB) / 256, 256, 0, stream>>>(x, embt, rnn_in);
    attn_ctx_kernel<<<BATCH,            256, 0, stream>>>(enc, hidden, W_e, b_e, rnn_in);
    gates_kernel   <<<G4 / 16,          256, 0, stream>>>(rnn_in, hidden, W_ih, W_hh,
                                                          b_ih, b_hh, gates);
    lstm_kernel    <<<(BATCH * HID) / 256, 256, 0, stream>>>(gates, cell, h_new, c_new);
    fc_kernel      <<<VOCAB / 16,       256, 0, stream>>>(h_new, W_fc, b_fc, preds);
}